// Model_46806553592159
// MI455X (gfx1250) — compile-verified
//
#include <hip/hip_runtime.h>
#include <hip/hip_bf16.h>

// ---------------------------------------------------------------------------
// GAT (2x GATConv + mean-pool + linear) for gfx1250 / MI455X.
// Dense transforms: V_WMMA_F32_16X16X4_F32 (fp32 WMMA, matches fp32 reference)
// with the A-tile staged in LDS (coalesced b128 global loads, bank-padded).
// Edge softmax/aggregation: channel-parallel coalesced float atomics.
// ---------------------------------------------------------------------------

#define NGRAPH  32
#define NCLS    5
#define NEG_SLOPE 0.2f

typedef __attribute__((ext_vector_type(2))) float v2f;
typedef __attribute__((ext_vector_type(8))) float v8f;

// ---------------------------------------------------------------------------
// C[M,Ncol] = A[M,KK] * B[KK,Ncol], row-major. blockDim = (32,4): 4 waves,
// each owning one 16x16 tile in a 16x64 block tile. The 16xKK A-tile is
// staged once per block in LDS (padded stride KK+4 -> conflict-free column
// reads, float4-aligned). KK is compile-time (128 / 256), KK % 4 == 0.
// WMMA f32 16x16x4 lane layout (ISA 7.12.2):
//   A: lanes 0-15 hold K=k,k+1 ; lanes 16-31 hold K=k+2,k+3 (for row M=lane%16)
//   B: VGPR r -> row k+r+2*hi (col = lane%16)
//   C/D: VGPR r -> row (r + 8*hi), col = lane%16
// ---------------------------------------------------------------------------
template <int KK>
__global__ void gemm_wmma_f32(const float* __restrict__ A,
                              const float* __restrict__ B,
                              float* __restrict__ C,
                              int M, int Ncol) {
    constexpr int LDA = KK + 4;               // pad: stride mod 64 == 4 banks
    constexpr int CH4 = KK / 4;               // float4s per row
    __shared__ float As[16 * LDA];

    const int lane = threadIdx.x;             // 0..31
    const int lo   = lane & 15;
    const int hi   = lane >> 4;               // 0 or 1
    const int m0   = blockIdx.x * 16;
    const int n0   = blockIdx.y * 64 + threadIdx.y * 16;
    const int tid  = threadIdx.y * 32 + threadIdx.x;   // 0..127

    // cooperative, coalesced load of the 16 x KK A-tile (global b128)
    for (int idx = tid; idx < 16 * CH4; idx += 128) {
        const int r = idx / CH4;
        const int c = idx % CH4;
        const float4 v = *(const float4*)(A + (size_t)(m0 + r) * KK + 4 * c);
        *(float4*)&As[r * LDA + 4 * c] = v;
    }
    __syncthreads();

    v8f acc = {};
    const float* arow = As + lo * LDA;
#pragma unroll 8
    for (int k = 0; k < KK; k += 4) {
        const int kk = k + 2 * hi;
        v2f a, b;
        a.x = arow[kk];                        // ds_load (bank-conflict free)
        a.y = arow[kk + 1];
        b.x = B[(kk)     * Ncol + n0 + lo];    // 64B coalesced per row, L2-hot
        b.y = B[(kk + 1) * Ncol + n0 + lo];
        acc = __builtin_amdgcn_wmma_f32_16x16x4_f32(
            /*neg_a=*/false, a, /*neg_b=*/false, b,
            /*c_mod=*/(short)0, acc, /*reuse_a=*/false, /*reuse_b=*/false);
    }
#pragma unroll
    for (int r = 0; r < 8; ++r) {
        C[(size_t)(m0 + r + 8 * hi) * Ncol + n0 + lo] = acc[r];
    }
}

// ---------------------------------------------------------------------------
__global__ void fill_zero(float* __restrict__ p, int n) {
    int i = blockIdx.x * blockDim.x + threadIdx.x;
    if (i < n) p[i] = 0.0f;
}

__device__ __forceinline__ float leaky(float x) {
    return x > 0.0f ? x : NEG_SLOPE * x;
}
// order-preserving uint encoding of float (for atomicMax); enc(x) >= 0x007FFFFF
// for every finite x, so a 0-initialized buffer acts as -inf.
__device__ __forceinline__ unsigned int f2ord(float f) {
    unsigned int u = __float_as_uint(f);
    return (u & 0x80000000u) ? ~u : (u | 0x80000000u);
}
__device__ __forceinline__ float ord2f(unsigned int e) {
    unsigned int u = (e & 0x80000000u) ? (e ^ 0x80000000u) : ~e;
    return __uint_as_float(u);
}

// a_src[n,h] = sum_c h[n,h,c] * att_src[h,c]   (same for dst), float4 dot
template <int H, int Cc>
__global__ void attn_scores_t(const float* __restrict__ h,
                              const float* __restrict__ att_s,
                              const float* __restrict__ att_d,
                              float* __restrict__ asrc,
                              float* __restrict__ adst,
                              int n) {
    int i = blockIdx.x * blockDim.x + threadIdx.x;   // node*H + head
    if (i >= n * H) return;
    const int hd = i % H;
    const float4* hp = (const float4*)(h + (size_t)i * Cc);
    const float4* sp = (const float4*)(att_s + hd * Cc);
    const float4* dp = (const float4*)(att_d + hd * Cc);
    float s = 0.0f, d = 0.0f;
#pragma unroll 4
    for (int c = 0; c < Cc / 4; ++c) {
        const float4 v = hp[c], a = sp[c], b = dp[c];
        s += v.x * a.x + v.y * a.y + v.z * a.z + v.w * a.w;
        d += v.x * b.x + v.y * b.y + v.z * b.z + v.w * b.w;
    }
    asrc[i] = s;
    adst[i] = d;
}

// segment-max of leaky_relu(alpha) over dst
template <int H>
__global__ void edge_max_t(const int* __restrict__ src_i,
                           const int* __restrict__ dst_i,
                           int Ereal, int n,
                           const float* __restrict__ asrc,
                           const float* __restrict__ adst,
                           unsigned int* __restrict__ amax) {
    int i = blockIdx.x * blockDim.x + threadIdx.x;   // edge*H + head
    if (i >= (Ereal + n) * H) return;
    const int e = i / H, hd = i % H;
    const int s = (e < Ereal) ? src_i[e] : (e - Ereal);
    const int d = (e < Ereal) ? dst_i[e] : (e - Ereal);
    const float alpha = leaky(asrc[s * H + hd] + adst[d * H + hd]);
    atomicMax(&amax[d * H + hd], f2ord(alpha));
}

// fused softmax-weight + aggregation, channel-parallel:
// one thread per (edge, head, 4-channel group) -> coalesced b128 gather of
// h[src] and cache-line-coalesced atomic adds into aggr[dst].
template <int H, int Cc>
__global__ void edge_aggr_t(const int* __restrict__ src_i,
                            const int* __restrict__ dst_i,
                            int Ereal, int n,
                            const float* __restrict__ asrc,
                            const float* __restrict__ adst,
                            const unsigned int* __restrict__ amax,
                            const float* __restrict__ h,
                            float* __restrict__ denom,
                            float* __restrict__ aggr) {
    constexpr int C4  = Cc / 4;
    constexpr int CPE = H * C4;                      // threads per edge
    long long i = (long long)blockIdx.x * blockDim.x + threadIdx.x;
    if (i >= (long long)(Ereal + n) * CPE) return;
    const int e   = (int)(i / CPE);
    const int rem = (int)(i % CPE);
    const int hd  = rem / C4;
    const int c4  = rem % C4;
    const int s = (e < Ereal) ? src_i[e] : (e - Ereal);
    const int d = (e < Ereal) ? dst_i[e] : (e - Ereal);
    const float alpha = leaky(asrc[s * H + hd] + adst[d * H + hd]);
    const float w = expf(alpha - ord2f(amax[d * H + hd]));
    if (c4 == 0) atomicAdd(&denom[d * H + hd], w);
    const float4 hv = *(const float4*)(h + ((size_t)s * H + hd) * Cc + 4 * c4);
    float* op = aggr + ((size_t)d * H + hd) * Cc + 4 * c4;
    atomicAdd(op + 0, w * hv.x);
    atomicAdd(op + 1, w * hv.y);
    atomicAdd(op + 2, w * hv.z);
    atomicAdd(op + 3, w * hv.w);
}

// in-place: x = elu(aggr / (denom + eps) + bias)
template <int H, int Cc>
__global__ void finalize_elu_t(float* __restrict__ aggr,
                               const float* __restrict__ denom,
                               const float* __restrict__ bias,
                               int n) {
    int i = blockIdx.x * blockDim.x + threadIdx.x;
    if (i >= n * H * Cc) return;
    const int node = i / (H * Cc);
    const int rem  = i % (H * Cc);
    const int hd   = rem / Cc;
    const float v = aggr[i] / (denom[node * H + hd] + 1e-16f) + bias[rem];
    aggr[i] = v > 0.0f ? v : (expf(v) - 1.0f);
}

// mean-pool accumulation
__global__ void pool_sum(const float* __restrict__ h2,
                         const int* __restrict__ batch,
                         float* __restrict__ sums,
                         float* __restrict__ cnt,
                         int n, int Cc) {
    int i = blockIdx.x * blockDim.x + threadIdx.x;
    if (i >= n * Cc) return;
    const int node = i / Cc, c = i % Cc;
    const int g = batch[node];
    atomicAdd(&sums[g * Cc + c], h2[i]);
    if (c == 0) atomicAdd(&cnt[g], 1.0f);
}

// out[g,k] = (sums[g,:]/max(cnt,1)) @ lin_w[:,k] + lin_b[k]
__global__ void head_linear(const float* __restrict__ sums,
                            const float* __restrict__ cnt,
                            const float* __restrict__ lw,
                            const float* __restrict__ lb,
                            float* __restrict__ out,
                            int G, int Cc, int CLS) {
    int i = blockIdx.x * blockDim.x + threadIdx.x;
    if (i >= G * CLS) return;
    const int g = i / CLS, k = i % CLS;
    const float inv = 1.0f / fmaxf(cnt[g], 1.0f);
    float acc = lb[k];
    for (int c = 0; c < Cc; ++c)
        acc += sums[g * Cc + c] * inv * lw[c * CLS + k];
    out[i] = acc;
}

// ---------------------------------------------------------------------------
static inline int ceildiv(long long a, int b) { return (int)((a + b - 1) / b); }

extern "C" void kernel_launch(void* const* d_in, const int* in_sizes, int n_in,
                              void* d_out, int out_size, void* d_ws, size_t ws_size,
                              hipStream_t stream) {
    const float* x        = (const float*)d_in[0];
    const int*   eidx     = (const int*)d_in[1];     // [2,E]
    const int*   batch    = (const int*)d_in[2];     // [N]
    const float* W1       = (const float*)d_in[3];
    const float* att_src1 = (const float*)d_in[4];
    const float* att_dst1 = (const float*)d_in[5];
    const float* b1       = (const float*)d_in[6];
    const float* W2       = (const float*)d_in[7];
    const float* att_src2 = (const float*)d_in[8];
    const float* att_dst2 = (const float*)d_in[9];
    const float* b2       = (const float*)d_in[10];
    const float* lin_w    = (const float*)d_in[11];
    const float* lin_b    = (const float*)d_in[12];
    float* out = (float*)d_out;

    const int E  = in_sizes[1] / 2;                  // 800000
    const int N  = in_sizes[2];                      // 50000 (16 | N)
    const int F1 = 256;                              // H1*C1
    const int* src_i = eidx;
    const int* dst_i = eidx + E;

    // workspace layout (floats)
    float* ws = (float*)d_ws;
    float*        hbuf  = ws;                                    // N*256 (h; reused layer2)
    float*        aggr1 = hbuf  + (size_t)N * 256;               // N*256 -> x2 after elu
    float*        aggr2 = aggr1 + (size_t)N * 256;               // N*128
    float*        asrc  = aggr2 + (size_t)N * 128;               // N*4
    float*        adst  = asrc  + (size_t)N * 4;                 // N*4
    float*        amaxf = adst  + (size_t)N * 4;                 // N*4 (uint-encoded)
    float*        den   = amaxf + (size_t)N * 4;                 // N*4
    float*        psum  = den   + (size_t)N * 4;                 // 32*128
    float*        pcnt  = psum  + (size_t)NGRAPH * 128;          // 32
    unsigned int* amax  = (unsigned int*)amaxf;

    const int TB = 256;
    const dim3 wblk(32, 4);                                      // 4 waves / block

    // ================= Layer 1 (H=4, C=64) =================
    fill_zero<<<ceildiv(N * 256, TB), TB, 0, stream>>>(aggr1, N * 256);
    fill_zero<<<ceildiv(N * 8, TB), TB, 0, stream>>>(amaxf, N * 8);   // amax+den

    gemm_wmma_f32<128><<<dim3(N / 16, 256 / 64), wblk, 0, stream>>>(x, W1, hbuf, N, 256);

    attn_scores_t<4, 64><<<ceildiv(N * 4, TB), TB, 0, stream>>>(hbuf, att_src1, att_dst1,
                                                                asrc, adst, N);
    edge_max_t<4><<<ceildiv((long long)(E + N) * 4, TB), TB, 0, stream>>>(
        src_i, dst_i, E, N, asrc, adst, amax);
    edge_aggr_t<4, 64><<<ceildiv((long long)(E + N) * 64, TB), TB, 0, stream>>>(
        src_i, dst_i, E, N, asrc, adst, amax, hbuf, den, aggr1);
    finalize_elu_t<4, 64><<<ceildiv(N * 256, TB), TB, 0, stream>>>(aggr1, den, b1, N);

    // ================= Layer 2 (H=1, C=128) =================
    fill_zero<<<ceildiv(N * 128, TB), TB, 0, stream>>>(aggr2, N * 128);
    fill_zero<<<ceildiv(N * 8, TB), TB, 0, stream>>>(amaxf, N * 8);

    gemm_wmma_f32<256><<<dim3(N / 16, 128 / 64), wblk, 0, stream>>>(aggr1, W2, hbuf, N, 128);

    attn_scores_t<1, 128><<<ceildiv(N, TB), TB, 0, stream>>>(hbuf, att_src2, att_dst2,
                                                             asrc, adst, N);
    edge_max_t<1><<<ceildiv((long long)(E + N), TB), TB, 0, stream>>>(
        src_i, dst_i, E, N, asrc, adst, amax);
    edge_aggr_t<1, 128><<<ceildiv((long long)(E + N) * 32, TB), TB, 0, stream>>>(
        src_i, dst_i, E, N, asrc, adst, amax, hbuf, den, aggr2);
    finalize_elu_t<1, 128><<<ceildiv(N * 128, TB), TB, 0, stream>>>(aggr2, den, b2, N);

    // ================= Pool + head =================
    fill_zero<<<ceildiv(NGRAPH * 128 + NGRAPH, TB), TB, 0, stream>>>(psum, NGRAPH * 128 + NGRAPH);
    pool_sum<<<ceildiv(N * 128, TB), TB, 0, stream>>>(aggr2, batch, psum, pcnt, N, 128);
    head_linear<<<1, TB, 0, stream>>>(psum, pcnt, lin_w, lin_b, out, NGRAPH, 128, NCLS);
}